// SelfAttention_42898133353002
// MI455X (gfx1250) — compile-verified
//
#include <hip/hip_runtime.h>
#include <stdint.h>

// B=4, S=2048, D_IN=1024, H=1024, 3H=3072
#define S_LEN 2048
#define DIN   1024
#define HDIM  1024
#define N3H   3072

typedef __attribute__((ext_vector_type(2))) float v2f;
typedef __attribute__((ext_vector_type(8))) float v8f;

// D = A(16x4,f32) * B(4x16,f32) + C(16x16,f32)   -- V_WMMA_F32_16X16X4_F32
__device__ inline v8f wmma4(v2f a, v2f b, v8f c) {
    return __builtin_amdgcn_wmma_f32_16x16x4_f32(
        /*neg_a=*/false, a, /*neg_b=*/false, b,
        /*c_mod=*/(short)0, c, /*reuse_a=*/false, /*reuse_b=*/false);
}

// Async DMA: 16 bytes per lane, global -> LDS. Tracked by ASYNCcnt.
// lds_off = LDS byte offset (low 32 bits of flat shared pointer), per lane.
__device__ inline void async_v_load_b128(uint32_t lds_off, const float* gsrc) {
    asm volatile("global_load_async_to_lds_b128 %0, %1, off"
                 :: "v"(lds_off), "v"(gsrc)
                 : "memory");
}
__device__ inline void wait_async_all() {
    asm volatile("s_wait_asynccnt 0x0" ::: "memory");
}

// ---------------------------------------------------------------------------
// Kernel 1: proj = qkv @ W + b  (M=8192, K=1024, N=3072), Q columns scaled x32.
// Grid: (M/16, N/512). Block = 256 threads = 8 waves; wave w does cols
// [blockIdx.y*512 + w*64, +64) as 4 accumulators sharing one A fragment.
// ---------------------------------------------------------------------------
__global__ __launch_bounds__(256) void qkv_proj_kernel(
    const float* __restrict__ A, const float* __restrict__ W,
    const float* __restrict__ bias, float* __restrict__ proj) {
    const int tid   = threadIdx.x;
    const int lane  = tid & 31;
    const int wave  = tid >> 5;
    const int l15   = lane & 15;
    const int khalf = (lane >> 4) << 1;   // 0 for lanes 0-15, 2 for 16-31
    const int rowhi = (lane >> 4) * 8;    // C/D row offset for upper half-wave

    const int m0 = blockIdx.x * 16;
    const int n0 = blockIdx.y * 512 + wave * 64;

    v8f acc[4] = {};
    const float* arow = A + (size_t)(m0 + l15) * DIN + khalf;
    const float* wcol = W + n0 + l15;

    for (int k = 0; k < DIN; k += 4) {
        v2f a = *(const v2f*)(arow + k);               // A[m][k+khalf .. +1]
        const float* wp = wcol + (size_t)(k + khalf) * N3H;
        #pragma unroll
        for (int j = 0; j < 4; ++j) {
            v2f b;
            b.x = wp[16 * j];                          // W[k+khalf  ][n]
            b.y = wp[N3H + 16 * j];                    // W[k+khalf+1][n]
            acc[j] = wmma4(a, b, acc[j]);
        }
    }

    #pragma unroll
    for (int j = 0; j < 4; ++j) {
        const int col = n0 + 16 * j + l15;
        const float bv = bias[col];
        const float qs = (col < HDIM) ? 32.0f : 1.0f;  // sqrt(H)=32 folded into Q
        #pragma unroll
        for (int r = 0; r < 8; ++r) {
            const int row = m0 + r + rowhi;
            proj[(size_t)row * N3H + col] = (acc[j][r] + bv) * qs;
        }
    }
}

// ---------------------------------------------------------------------------
// Kernel 2: flash attention over proj. One block per (batch, 16-query tile).
// 8 waves; wave w owns output columns [w*128, w*128+128).
// Per 16-key tile:
//   - async-DMA this wave's 16x128 V slice into LDS (overlaps all of QK^T)
//   - partial QK^T per wave (f32 WMMA), ds_add_f32 reduction in LDS
//   - online softmax by 16 row-owner threads
//   - s_wait_asynccnt 0, then P*V WMMAs reading V fragments from LDS
// ---------------------------------------------------------------------------
__global__ __launch_bounds__(256) void flash_attn_kernel(
    const float* __restrict__ proj, const uint8_t* __restrict__ mask,
    float* __restrict__ out) {
    __shared__ __align__(16) float vbuf[8][16][128];  // 64 KB V staging (per-wave slices)
    __shared__ float s_red[256];                      // 16x16 score reduction
    __shared__ __align__(16) float p_tile[256];       // 16x16 P, row-major
    __shared__ float row_scale[16];
    __shared__ float row_linv[16];

    const int tid   = threadIdx.x;
    const int lane  = tid & 31;
    const int wave  = tid >> 5;
    const int l15   = lane & 15;
    const int khalf = (lane >> 4) << 1;
    const int rowhi = (lane >> 4) * 8;

    const int b     = blockIdx.x >> 7;               // / (S/16)
    const int q0    = (blockIdx.x & 127) * 16;
    const int hbase = wave * 128;

    const float*   base = proj + (size_t)b * S_LEN * N3H;
    const uint8_t* mb   = mask + (size_t)b * S_LEN;

    // LDS byte offset of this wave's V slice, per lane (16B chunk per lane).
    const uint32_t vlds0 = (uint32_t)(uintptr_t)&vbuf[wave][0][lane * 4];

    // Preload this wave's Q fragments (16 rows x 128 cols) into registers.
    v2f qf[32];
    {
        const float* qrow = base + (size_t)(q0 + l15) * N3H + hbase + khalf;
        #pragma unroll
        for (int f = 0; f < 32; ++f) qf[f] = *(const v2f*)(qrow + 4 * f);
    }

    v8f acc[8] = {};                                 // 16 rows x 128 cols of O
    float m_run = -3.0e38f;                          // row state (tid<16 only)
    float l_run = 0.0f;

    for (int k0 = 0; k0 < S_LEN; k0 += 16) {
        // Kick off async V-tile DMA for this key tile (own slice only; it
        // overlaps the whole QK^T + reduction + softmax below).
        {
            const float* vsrc = base + (size_t)k0 * N3H + 2 * HDIM + hbase + lane * 4;
            #pragma unroll
            for (int r = 0; r < 16; ++r)
                async_v_load_b128(vlds0 + r * 512, vsrc + (size_t)r * N3H);
        }

        __syncthreads();                             // prev p_tile/s_red consumed
        s_red[tid] = 0.0f;
        __syncthreads();

        // Partial scores over this wave's H-slice: 32 x WMMA f32 16x16x4.
        v8f c = {};
        const float* krow = base + (size_t)(k0 + l15) * N3H + HDIM + hbase + khalf;
        if (k0 + 16 < S_LEN)
            __builtin_prefetch(krow + 16 * N3H, 0, 0);  // next key tile K rows
        #pragma unroll
        for (int f = 0; f < 32; ++f) {
            v2f kf = *(const v2f*)(krow + 4 * f);
            c = wmma4(qf[f], kf, c);
        }
        #pragma unroll
        for (int r = 0; r < 8; ++r)
            atomicAdd(&s_red[(r + rowhi) * 16 + l15], c[r]);  // ds_add_f32
        __syncthreads();

        // Online softmax bookkeeping: thread t (<16) owns query row t.
        if (tid < 16) {
            const int row = tid;
            float sv[16];
            float mx = -3.0e38f;
            #pragma unroll
            for (int j = 0; j < 16; ++j) {
                float s = s_red[row * 16 + j];
                if (mb[k0 + j]) s = -3.0e38f;        // masked key
                sv[j] = s;
                mx = fmaxf(mx, s);
            }
            const float m_new = fmaxf(m_run, mx);
            const float scale = __expf(m_run - m_new);
            float ps = 0.0f;
            #pragma unroll
            for (int j = 0; j < 16; ++j) {
                float p = __expf(sv[j] - m_new);
                p_tile[row * 16 + j] = p;
                ps += p;
            }
            l_run = l_run * scale + ps;
            m_run = m_new;
            row_scale[row] = scale;
        }
        __syncthreads();

        // Rescale accumulators.
        #pragma unroll
        for (int r = 0; r < 8; ++r) {
            const float sc = row_scale[r + rowhi];
            #pragma unroll
            for (int t = 0; t < 8; ++t) acc[t][r] *= sc;
        }

        // O += P * V, with V fragments served from LDS (DMA'd above).
        wait_async_all();                            // s_wait_asynccnt 0
        #pragma unroll
        for (int kk = 0; kk < 4; ++kk) {
            v2f a = *(const v2f*)(&p_tile[l15 * 16 + 4 * kk + khalf]);
            const int vr = 4 * kk + khalf;
            #pragma unroll
            for (int t = 0; t < 8; ++t) {
                v2f bv;
                bv.x = vbuf[wave][vr][16 * t + l15];     // V[k  ][h] (ds_load)
                bv.y = vbuf[wave][vr + 1][16 * t + l15]; // V[k+1][h]
                acc[t] = wmma4(a, bv, acc[t]);
            }
        }
    }

    __syncthreads();
    if (tid < 16) row_linv[tid] = 1.0f / l_run;
    __syncthreads();

    float* orow = out + ((size_t)b * S_LEN + q0) * HDIM;
    #pragma unroll
    for (int r = 0; r < 8; ++r) {
        const int row = r + rowhi;
        const float li = row_linv[row];
        #pragma unroll
        for (int t = 0; t < 8; ++t)
            orow[(size_t)row * HDIM + hbase + 16 * t + l15] = acc[t][r] * li;
    }
}

// ---------------------------------------------------------------------------
extern "C" void kernel_launch(void* const* d_in, const int* in_sizes, int n_in,
                              void* d_out, int out_size, void* d_ws, size_t ws_size,
                              hipStream_t stream) {
    const float*   qkv  = (const float*)d_in[0];     // (B,S,D_IN) f32
    const uint8_t* mask = (const uint8_t*)d_in[1];   // (B,S) bool
    const float*   W    = (const float*)d_in[2];     // (D_IN,3H) f32
    const float*   bias = (const float*)d_in[3];     // (3H,) f32
    float*         out  = (float*)d_out;             // (B,S,H) f32
    float*         proj = (float*)d_ws;              // B*S*3H f32 = 96 MB scratch

    dim3 gridP(512, 6);                              // (B*S/16, 3H/512)
    qkv_proj_kernel<<<gridP, 256, 0, stream>>>(qkv, W, bias, proj);
    flash_attn_kernel<<<512, 256, 0, stream>>>(proj, mask, out);
}